// LinearEncoder_30382598651961
// MI455X (gfx1250) — compile-verified
//
#include <hip/hip_runtime.h>
#include <hip/hip_bf16.h>

typedef __attribute__((ext_vector_type(2))) float v2f;
typedef __attribute__((ext_vector_type(8))) float v8f;

#define FEAT_IN  256
#define FEAT_OUT 64

// ---------------- degree / norm kernels ----------------

__global__ void deg_init_kernel(float* __restrict__ deg, int n) {
    int i = blockIdx.x * blockDim.x + threadIdx.x;
    if (i < n) deg[i] = 1.0f;   // self-loop contributes 1
}

__global__ void deg_accum_kernel(const long long* __restrict__ ei, float* __restrict__ deg,
                                 long long E) {
    long long e = (long long)blockIdx.x * blockDim.x + threadIdx.x;
    if (e < E) {
        long long col = ei[E + e];          // edge_index[1][e]
        atomicAdd(&deg[col], 1.0f);
    }
}

__global__ void dinv_kernel(const float* __restrict__ deg, float* __restrict__ dinv, int n) {
    int i = blockIdx.x * blockDim.x + threadIdx.x;
    if (i < n) dinv[i] = rsqrtf(deg[i]);    // deg >= 1 always (self-loop)
}

// ---------------- GEMM: h = x @ W via V_WMMA_F32_16X16X4_F32 ----------------
// Each wave owns a 16-row M tile and ALL 64 output columns: one A fragment is
// reused by 4 WMMAs per K-step (4 accumulators), so each row of x is fetched
// from memory exactly once. W (64 KB) is re-read by every block but is
// L0/L2-resident. 8 waves per block; tile guard is wave-uniform so EXEC is
// all-1s at every WMMA (ISA requirement).
//
// A (16x4 f32) per-lane: lanes 0-15 hold row M=lane, K={0,1}; lanes 16-31 hold
// row M=lane-16, K={2,3}. B (4x16) mirrored: lane holds col N=lane%16, same K
// pair. C/D (16x16 f32): VGPR r, lanes 0-15 -> (M=r, N=lane); lanes 16-31 ->
// (M=r+8, N=lane-16).

__global__ void __launch_bounds__(256)
gemm_wmma_kernel(const float* __restrict__ x, const float* __restrict__ W,
                 float* __restrict__ h, int n) {
    const int lane = threadIdx.x & 31;
    const int wave = threadIdx.x >> 5;
    const int tile = blockIdx.x * 8 + wave;     // 16-row M tile index
    const int m0   = tile * 16;
    if (m0 >= n) return;                        // wave-uniform guard
    const int t  = lane & 15;                   // row (A) / col (B) within tile
    const int kh = (lane >> 4) * 2;             // K sub-offset: 0 or 2

    v8f acc0 = {0.f,0.f,0.f,0.f,0.f,0.f,0.f,0.f};
    v8f acc1 = acc0, acc2 = acc0, acc3 = acc0;

    const float* xrow = x + (long long)(m0 + t) * FEAT_IN;
    const float* wcol = W + t;

    #pragma unroll 2
    for (int k0 = 0; k0 < FEAT_IN; k0 += 4) {
        v2f a;
        a.x = xrow[k0 + kh];                    // 8B-aligned contiguous pair -> b64
        a.y = xrow[k0 + kh + 1];
        const float* wp = wcol + (k0 + kh) * FEAT_OUT;
        v2f b0, b1, b2, b3;
        b0.x = wp[0];  b0.y = wp[FEAT_OUT + 0];
        b1.x = wp[16]; b1.y = wp[FEAT_OUT + 16];
        b2.x = wp[32]; b2.y = wp[FEAT_OUT + 32];
        b3.x = wp[48]; b3.y = wp[FEAT_OUT + 48];
        acc0 = __builtin_amdgcn_wmma_f32_16x16x4_f32(false, a, false, b0, (short)0, acc0, false, false);
        acc1 = __builtin_amdgcn_wmma_f32_16x16x4_f32(false, a, false, b1, (short)0, acc1, false, false);
        acc2 = __builtin_amdgcn_wmma_f32_16x16x4_f32(false, a, false, b2, (short)0, acc2, false, false);
        acc3 = __builtin_amdgcn_wmma_f32_16x16x4_f32(false, a, false, b3, (short)0, acc3, false, false);
    }

    const int nn    = lane & 15;
    const int mbase = (lane >> 4) * 8;
    float* hrow = h + (long long)(m0 + mbase) * FEAT_OUT + nn;
    #pragma unroll
    for (int r = 0; r < 8; ++r) {
        long long off = (long long)r * FEAT_OUT;
        hrow[off +  0] = acc0[r];
        hrow[off + 16] = acc1[r];
        hrow[off + 32] = acc2[r];
        hrow[off + 48] = acc3[r];
    }
}

// ---------------- out init: self-loop term + bias (overwrites poisoned d_out) ----

__global__ void out_init_kernel(const float* __restrict__ h, const float* __restrict__ dinv,
                                const float* __restrict__ b, float* __restrict__ out, int n) {
    long long idx = (long long)blockIdx.x * blockDim.x + threadIdx.x;
    long long total = (long long)n * FEAT_OUT;
    if (idx < total) {
        long long node = idx >> 6;           // FEAT_OUT == 64
        int c = (int)(idx & 63);
        float di = dinv[node];
        out[idx] = h[idx] * di * di + b[c];  // self-loop norm = dinv^2
    }
}

// ---------------- edge scatter: out[col] += h[row] * dinv[row]*dinv[col] --------
// thread = (edge, channel): 64 consecutive lanes share an edge -> coalesced
// gather loads and coalesced global_atomic_add_f32 into out. Whole working set
// (~180 MB) fits the 192 MB L2, so random node access runs at L2 bandwidth.

__global__ void edge_scatter_kernel(const long long* __restrict__ ei,
                                    const float* __restrict__ h,
                                    const float* __restrict__ dinv,
                                    float* __restrict__ out, long long E) {
    long long tid = (long long)blockIdx.x * blockDim.x + threadIdx.x;
    long long e = tid >> 6;
    if (e >= E) return;
    int c = (int)(tid & 63);
    long long row = ei[e];                   // source
    long long col = ei[E + e];               // target
    float nm = dinv[row] * dinv[col];
    float v  = h[row * FEAT_OUT + c] * nm;
    atomicAdd(&out[col * FEAT_OUT + c], v);
}

// ---------------- launcher ----------------

extern "C" void kernel_launch(void* const* d_in, const int* in_sizes, int n_in,
                              void* d_out, int out_size, void* d_ws, size_t ws_size,
                              hipStream_t stream) {
    const float*     x  = (const float*)d_in[0];
    const long long* ei = (const long long*)d_in[1];   // int64 [2, E] flat
    const float*     W  = (const float*)d_in[2];
    const float*     b  = (const float*)d_in[3];
    float* out = (float*)d_out;

    const int       n = in_sizes[0] / FEAT_IN;         // 100000
    const long long E = (long long)in_sizes[1] / 2;    // 1600000

    // workspace layout: h [n*64 f32] | deg [n f32] | dinv [n f32]
    char* ws = (char*)d_ws;
    float* h    = (float*)ws;
    float* deg  = (float*)(ws + (((size_t)n * FEAT_OUT * sizeof(float) + 255) & ~(size_t)255));
    float* dinv = deg + n;

    const int B = 256;

    deg_init_kernel<<<(n + B - 1) / B, B, 0, stream>>>(deg, n);
    deg_accum_kernel<<<(int)((E + B - 1) / B), B, 0, stream>>>(ei, deg, E);
    dinv_kernel<<<(n + B - 1) / B, B, 0, stream>>>(deg, dinv, n);

    const int tiles = (n + 15) / 16;                   // 6250
    gemm_wmma_kernel<<<(tiles + 7) / 8, 256, 0, stream>>>(x, W, h, n);

    long long total = (long long)n * FEAT_OUT;
    out_init_kernel<<<(int)((total + B - 1) / B), B, 0, stream>>>(h, dinv, b, out, n);

    long long work = E * FEAT_OUT;
    edge_scatter_kernel<<<(int)((work + B - 1) / B), B, 0, stream>>>(ei, h, dinv, out, E);
}